// AnatomicalTextEnhancer_16157666968243
// MI455X (gfx1250) — compile-verified
//
#include <hip/hip_runtime.h>

#define B_  16
#define R_  29
#define H_  768
#define N_  8000
#define K_  5
#define QS_STRIDE 772   // 768 + 4 pad -> bank-conflict-free ds_load_b64

typedef float v2f __attribute__((ext_vector_type(2)));
typedef float v8f __attribute__((ext_vector_type(8)));

// ---------------------------------------------------------------------------
// Kernel 1: L2-normalize query rows. One 256-thread block per (b,r) row.
// ---------------------------------------------------------------------------
__global__ __launch_bounds__(256) void qnorm_kernel(const float* __restrict__ q,
                                                    float* __restrict__ qn) {
    const int row = blockIdx.x;                 // b*R_ + r
    const float* src = q + (size_t)row * H_;
    float s = 0.f;
    for (int h = threadIdx.x; h < H_; h += 256) {
        float v = src[h];
        s += v * v;
    }
    for (int o = 16; o > 0; o >>= 1) s += __shfl_xor(s, o, 32);
    __shared__ float red[8];
    __shared__ float inv_s;
    if ((threadIdx.x & 31) == 0) red[threadIdx.x >> 5] = s;
    __syncthreads();
    if (threadIdx.x == 0) {
        float t = 0.f;
        #pragma unroll
        for (int i = 0; i < 8; ++i) t += red[i];
        inv_s = 1.0f / fmaxf(sqrtf(t), 1e-12f);
    }
    __syncthreads();
    float inv = inv_s;
    for (int h = threadIdx.x; h < H_; h += 256)
        qn[(size_t)row * H_ + h] = src[h] * inv;
}

// ---------------------------------------------------------------------------
// Kernel 2: sims[b,r,n] = <q̂, bank_row>/max(||bank_row||,eps) via
// V_WMMA_F32_16X16X4_F32.  Grid (125, 29), 128 threads = 4 waves,
// each wave owns one 16-row bank tile; K loop = 768/4 = 192 WMMAs.
// ---------------------------------------------------------------------------
__global__ __launch_bounds__(128) void sims_wmma_kernel(const float* __restrict__ qn,
                                                        const float* __restrict__ bank,
                                                        float* __restrict__ sims) {
    __shared__ float Qs[16 * QS_STRIDE];

    const int r = blockIdx.y;

    // Stage the (16 x 768) normalized query slab for this r into LDS.
    for (int i = threadIdx.x; i < 16 * H_; i += 128) {
        const int b = i / H_;
        const int h = i - b * H_;
        Qs[b * QS_STRIDE + h] = qn[((size_t)b * R_ + r) * H_ + h];
    }
    __syncthreads();

    const int wave = threadIdx.x >> 5;
    const int lane = threadIdx.x & 31;
    const int col  = lane & 15;        // M (for A) / N (for B)
    const int hi   = lane >> 4;        // K-half selector

    const int n0   = (blockIdx.x * 4 + wave) * 16;
    const int nrow = n0 + col;

    const float* bp = bank + ((size_t)r * N_ + nrow) * H_ + 2 * hi;
    const float* qp = &Qs[col * QS_STRIDE + 2 * hi];

    v8f acc = {0.f, 0.f, 0.f, 0.f, 0.f, 0.f, 0.f, 0.f};
    float nrm = 0.f;

    #pragma unroll 4
    for (int k = 0; k < H_; k += 4) {
        v2f a = *(const v2f*)(qp + k);   // A: Q[m][k+2hi .. +1]   (LDS, b64)
        v2f b = *(const v2f*)(bp + k);   // B: bank[n][k+2hi .. +1] (global, b64)
        nrm += b.x * b.x + b.y * b.y;    // fused row-norm accumulation
        acc = __builtin_amdgcn_wmma_f32_16x16x4_f32(
            /*neg_a=*/false, a, /*neg_b=*/false, b,
            /*c_mod=*/(short)0, acc, /*reuse_a=*/false, /*reuse_b=*/false);
    }

    // lanes L and L+16 hold complementary K-halves of the same bank row
    nrm += __shfl_xor(nrm, 16, 32);
    const float inv = 1.0f / fmaxf(sqrtf(nrm), 1e-12f);

    // C/D layout: VGPR j, lanes 0-15 -> M=j, lanes 16-31 -> M=j+8; N = col.
    #pragma unroll
    for (int j = 0; j < 8; ++j) {
        const int m = j + 8 * hi;
        sims[((size_t)m * R_ + r) * N_ + n0 + col] = acc[j] * inv;
    }
}

// ---------------------------------------------------------------------------
// Kernel 3: masked top-5 per (b,r) row. One 256-thread block per row.
// ---------------------------------------------------------------------------
__global__ __launch_bounds__(256) void topk_kernel(const float* __restrict__ sims,
                                                   const int* __restrict__ excl,
                                                   float* __restrict__ out_sims,
                                                   float* __restrict__ out_idx,
                                                   float* __restrict__ out_best) {
    const int row = blockIdx.x;                 // b*R_ + r
    const float* s = sims + (size_t)row * N_;
    const int ex = excl[row];
    const int t = threadIdx.x;
    const float NEG_INF = -__builtin_inff();

    float v[K_];
    int   id[K_];
    #pragma unroll
    for (int i = 0; i < K_; ++i) { v[i] = NEG_INF; id[i] = 0x7fffffff; }

    for (int n = t; n < N_; n += 256) {
        const float val = (n == ex) ? NEG_INF : s[n];
        if (val > v[K_ - 1] || (val == v[K_ - 1] && n < id[K_ - 1])) {
            v[K_ - 1] = val; id[K_ - 1] = n;
            #pragma unroll
            for (int i = K_ - 1; i > 0; --i) {
                if (v[i] > v[i - 1] || (v[i] == v[i - 1] && id[i] < id[i - 1])) {
                    float tv = v[i]; v[i] = v[i - 1]; v[i - 1] = tv;
                    int   ti = id[i]; id[i] = id[i - 1]; id[i - 1] = ti;
                }
            }
        }
    }

    __shared__ float sv[256][K_];
    __shared__ int   si[256][K_];
    #pragma unroll
    for (int i = 0; i < K_; ++i) { sv[t][i] = v[i]; si[t][i] = id[i]; }
    __syncthreads();

    // log2(256) pairwise merges of sorted 5-lists
    for (int stride = 128; stride >= 1; stride >>= 1) {
        if (t < stride) {
            float mv[K_]; int mi[K_];
            int p = 0, q = 0;
            #pragma unroll
            for (int i = 0; i < K_; ++i) {          // p+q == i <= 4, in-bounds
                const float va = sv[t][p],       vb = sv[t + stride][q];
                const int   ia = si[t][p],       ib = si[t + stride][q];
                const bool takeA = (va > vb) || (va == vb && ia <= ib);
                if (takeA) { mv[i] = va; mi[i] = ia; ++p; }
                else       { mv[i] = vb; mi[i] = ib; ++q; }
            }
            #pragma unroll
            for (int i = 0; i < K_; ++i) { sv[t][i] = mv[i]; si[t][i] = mi[i]; }
        }
        __syncthreads();
    }

    if (t == 0) {
        #pragma unroll
        for (int i = 0; i < K_; ++i) {
            out_sims[row * K_ + i] = sv[0][i];
            out_idx [row * K_ + i] = (float)si[0][i];
        }
        out_best[row] = sv[0][0];
    }
}

// ---------------------------------------------------------------------------
extern "C" void kernel_launch(void* const* d_in, const int* in_sizes, int n_in,
                              void* d_out, int out_size, void* d_ws, size_t ws_size,
                              hipStream_t stream) {
    const float* query = (const float*)d_in[0];   // (16, 29, 768) f32
    const float* bank  = (const float*)d_in[1];   // (29, 8000, 768) f32
    const int*   excl  = (const int*)d_in[2];     // (16, 29) i32
    // d_in[3] = top_k scalar == 5 (compiled in)

    float* qn   = (float*)d_ws;                         // 16*29*768 floats
    float* sims = qn + (size_t)B_ * R_ * H_;            // 16*29*8000 floats

    float* out       = (float*)d_out;
    float* out_sims  = out;                             // 16*29*5
    float* out_idx   = out + B_ * R_ * K_;              // 16*29*5 (as float)
    float* out_best  = out + 2 * B_ * R_ * K_;          // 16*29

    qnorm_kernel<<<B_ * R_, 256, 0, stream>>>(query, qn);

    dim3 grid(N_ / 64, R_);                             // 125 x 29, 4 waves/WG
    sims_wmma_kernel<<<grid, 128, 0, stream>>>(qn, bank, sims);

    topk_kernel<<<B_ * R_, 256, 0, stream>>>(sims, excl, out_sims, out_idx, out_best);
}